// MelSpectrogramCNN_48722109006068
// MI455X (gfx1250) — compile-verified
//
#include <hip/hip_runtime.h>

// ============================================================================
// MI455X (gfx1250) Morlet scattering-2D + MLP head.
//
// Roofline reasoning (23.3 TB/s HBM, 192 MB L2, 320 KB LDS/WGP, wave32,
// WMMA-only matrix ops):
//  * MLP layer 1 streams w1 = 106752x1024 fp32 = 437 MB -> ~19 us floor at
//    HBM peak; M=16 batch makes it pure weight bandwidth. Keep fp32 weights
//    and run through V_WMMA_F32_16X16X4_F32 with a 16-way K-split.
//  * Every fft2/ifft2 of the scattering transform is a dense DFT-as-GEMM on
//    the f32 WMMA pipe (Y = X * W_n per axis, LDS-tiled transpose between
//    the two axis passes). Intermediates (<= 67 MB) fit the 192 MB L2.
//  * Rows [0, n/4) of the DFT-matrix B-panel (n x 16 complex) are staged into LDS once
//    per workgroup with GLOBAL_LOAD_ASYNC_TO_LDS_B128 (ASYNCcnt +
//    s_wait_asynccnt), shared by 8 M-tiles -> 8x less global W traffic.
//  * Morlet psi banks / Gaussian phi / DFT matrices are Python-module
//    constants, so they are regenerated deterministically on device.
// ============================================================================

#define PI_F     3.14159265358979323846f
#define TWOPI_F  6.28318530717958647693f
#define BN_EPS   1e-5f
#define BATCH    16
#define NCHAN    417
#define FEAT     (NCHAN * 256)   // 106752

typedef __attribute__((ext_vector_type(2))) float v2f;
typedef __attribute__((ext_vector_type(8))) float v8f;

__device__ __forceinline__ v8f wmma_f32x4(v2f a, v2f b, v8f c) {
  // V_WMMA_F32_16X16X4_F32 : D = A(16x4) * B(4x16) + C(16x16), fp32
  return __builtin_amdgcn_wmma_f32_16x16x4_f32(
      /*neg_a=*/false, a, /*neg_b=*/false, b,
      /*c_mod=*/(short)0, c, /*reuse_a=*/false, /*reuse_b=*/false);
}

// ------------------------------------------------------------ filter builders
__device__ __forceinline__ float fftfreq2pi(int p, int n) {
  int v = (p < (n >> 1)) ? p : (p - n);       // numpy fftfreq (even n)
  return TWOPI_F * (float)v / (float)n;
}

__global__ void fill_f(float* p, float v, int n) {
  int i = blockIdx.x * blockDim.x + threadIdx.x;
  if (i < n) p[i] = v;
}

__global__ void gen_dft(float2* W, int n, float sign, float scale) {
  int idx = blockIdx.x * blockDim.x + threadIdx.x;
  if (idx >= n * n) return;
  int j = idx / n, k = idx % n;
  int t = (int)(((long long)j * k) % n);
  float ang = sign * TWOPI_F * (float)t / (float)n;
  W[idx] = make_float2(scale * cosf(ang), scale * sinf(ang));
}

__global__ void gen_phi(float* phi, int n, int j) {
  int idx = blockIdx.x * blockDim.x + threadIdx.x;
  if (idx >= n * n) return;
  float kx = fftfreq2pi(idx / n, n);
  float ky = fftfreq2pi(idx % n, n);
  float sg = 0.8f * (float)(1 << j);
  phi[idx] = expf(-0.5f * sg * sg * (kx * kx + ky * ky));
}

__global__ void gen_psi_raw(float* psi, float* mx, int n, int j) {
  long long idx = (long long)blockIdx.x * blockDim.x + threadIdx.x;
  long long n2 = (long long)n * n;
  if (idx >= 8 * n2) return;
  int l = (int)(idx / n2);
  int p = (int)(idx % n2);
  float kx = fftfreq2pi(p / n, n);
  float ky = fftfreq2pi(p % n, n);
  float sg = 0.8f * (float)(1 << j);
  float xi = (3.0f * PI_F / 4.0f) / (float)(1 << j);
  float th = (float)l * PI_F / 8.0f;
  float ct = cosf(th), st = sinf(th);
  float w1 = kx * ct + ky * st;
  float w2 = -kx * st + ky * ct;
  float hs = 0.5f * sg * sg;
  float a   = expf(-hs * ((w1 - xi) * (w1 - xi) + 0.25f * w2 * w2));
  float env = expf(-hs * (w1 * w1 + 0.25f * w2 * w2));
  float beta = expf(-hs * xi * xi);
  float f = a - beta * env;
  psi[idx] = f;
  if (f > 0.0f) atomicMax((int*)&mx[l], __float_as_int(f)); // positive floats order as ints
}

__global__ void psi_norm(float* psi, const float* mx, int n2) {
  long long idx = (long long)blockIdx.x * blockDim.x + threadIdx.x;
  if (idx >= 8LL * n2) return;
  psi[idx] /= mx[idx / n2];
}

// ---------------------------------------------------------------- data movers
__global__ void r2c(float2* out, const float* in, long long total) {
  long long i = (long long)blockIdx.x * blockDim.x + threadIdx.x;
  if (i < total) out[i] = make_float2(in[i], 0.0f);
}

// out[o=(g,p)] = in[expand ? (g/nf, p) : o] * filt[(g%nf)*n2 + p]
__global__ void cmul_bcast(float2* out, const float2* in, const float* filt,
                           long long total, int n2, int nf, int expand) {
  long long o = (long long)blockIdx.x * blockDim.x + threadIdx.x;
  if (o >= total) return;
  int p = (int)(o % n2);
  long long g = o / n2;
  int f = (int)(g % nf);
  float2 v = expand ? in[(g / nf) * n2 + p] : in[o];
  float s = filt[(long long)f * n2 + p];
  out[o] = make_float2(v.x * s, v.y * s);
}

// out[g, i, j] = { |in[g, i*k, j*k]| , 0 }   (m = n/k)
__global__ void cmod_sub(float2* out, const float2* in, int n, int k, long long G) {
  int m = n / k;
  long long total = G * (long long)m * m;
  long long idx = (long long)blockIdx.x * blockDim.x + threadIdx.x;
  if (idx >= total) return;
  int p = (int)(idx % ((long long)m * m));
  long long g = idx / ((long long)m * m);
  int i = p / m, q = p % m;
  float2 v = in[g * (long long)n * n + (long long)(i * k) * n + q * k];
  out[idx] = make_float2(sqrtf(v.x * v.x + v.y * v.y), 0.0f);
}

// h[b, chanBase + cg*chanStride, 16x16] = real(in[g=(b,cg), ::k, ::k])
__global__ void real_sub_store(float* h, const float2* in, int n, int k,
                               int G, int Cg, int chanBase, int chanStride) {
  int idx = blockIdx.x * blockDim.x + threadIdx.x;
  if (idx >= G * 256) return;
  int p = idx & 255;
  int g = idx >> 8;
  int b = g / Cg, cg = g % Cg;
  int i = p >> 4, q = p & 15;
  float v = in[(long long)g * n * n + (long long)(i * k) * n + q * k].x;
  h[(long long)b * FEAT + (long long)(chanBase + cg * chanStride) * 256 + p] = v;
}

// batched n x n complex transpose via LDS tiles
__global__ void ctranspose(float2* __restrict__ out, const float2* __restrict__ in, int n) {
  __shared__ float2 t[16][17];
  long long base = (long long)blockIdx.z * n * n;
  int x = blockIdx.x * 16 + threadIdx.x;
  int y = blockIdx.y * 16 + threadIdx.y;
  t[threadIdx.y][threadIdx.x] = in[base + (long long)y * n + x];
  __syncthreads();
  int x2 = blockIdx.y * 16 + threadIdx.x;
  int y2 = blockIdx.x * 16 + threadIdx.y;
  out[base + (long long)y2 * n + x2] = t[threadIdx.x][threadIdx.y];
}

// ---------------------------------------------- complex GEMM: C = A * W (DFT)
// A: [Mrows x n] complex row-major, W: [n x n] complex DFT matrix.
// Workgroup = 8 waves sharing ONE N-column tile (tn); waves cover 8 M-tiles.
// Rows [0, n/4) of the W panel W[:, tn*16 .. tn*16+16) (n x 16 complex) are staged in
// LDS via the gfx1250 async global->LDS DMA path (ASYNCcnt), then all 8 waves
// consume it from LDS. 4 f32 WMMAs per K-step of 4; Ai negated in VALU
// (float WMMA NEG bits only permit C-negation per ISA).
__global__ __launch_bounds__(256) void cgemm_dft(float2* __restrict__ out,
                                                 const float2* __restrict__ A,
                                                 const float2* __restrict__ W,
                                                 int Mrows, int n) {
  __shared__ __align__(16) float2 sW[256 * 16];       // worst case n=256: 32 KB
  const int tn = blockIdx.x;
  const int lane = threadIdx.x & 31;
  const int wave = threadIdx.x >> 5;

  // ---- async-stage W panel: b128 transfer t = (row = t/8, colpair = t%8) ----
  const int nT = n * 2;                               // 2n b128 x 2 float2 = rows [0, n/4) of the panel
  for (int t = threadIdx.x; t < nT; t += 256) {
    int row = t >> 3;
    int cp  = t & 7;
    const float2* gp = W + (long long)row * n + tn * 16 + cp * 2;
    unsigned lds = (unsigned)(size_t)&sW[t * 2];      // low 32 bits = LDS offset
    asm volatile("global_load_async_to_lds_b128 %0, %1, off"
                 :: "v"(lds), "v"((unsigned long long)(size_t)gp)
                 : "memory");
  }
  asm volatile("s_wait_asynccnt 0x0" ::: "memory");   // drain this wave's DMA
  __syncthreads();                                    // panel visible WG-wide

  // ---- compute: one 16x16 complex tile per wave ----
  const int lr = lane & 15;
  const int hi = lane >> 4;
  const int tm = blockIdx.y * 8 + wave;
  const int row = tm * 16 + lr;                       // A fragment: M = lane&15
  const float2* __restrict__ Arow = A + (long long)row * n;
  v8f cr = {0.f, 0.f, 0.f, 0.f, 0.f, 0.f, 0.f, 0.f};
  v8f ci = {0.f, 0.f, 0.f, 0.f, 0.f, 0.f, 0.f, 0.f};
  for (int kk = 0; kk < n; kk += 4) {
    int k0 = kk + hi * 2;                             // K = {hi*2, hi*2+1}
    float2 a0 = Arow[k0];
    float2 a1 = Arow[k0 + 1];
    float2 b0 = sW[k0 * 16 + lr];                     // B fragments from LDS
    float2 b1 = sW[(k0 + 1) * 16 + lr];
    v2f ar  = { a0.x,  a1.x};
    v2f ai  = { a0.y,  a1.y};
    v2f mai = {-a0.y, -a1.y};
    v2f br  = { b0.x,  b1.x};
    v2f bi  = { b0.y,  b1.y};
    cr = wmma_f32x4(ar,  br, cr);   // + Ar*Br
    cr = wmma_f32x4(mai, bi, cr);   // - Ai*Bi
    ci = wmma_f32x4(ar,  bi, ci);   // + Ar*Bi
    ci = wmma_f32x4(ai,  br, ci);   // + Ai*Br
  }
  float2* Otile = out + (long long)(tm * 16) * n + tn * 16;
#pragma unroll
  for (int v = 0; v < 8; v++) {                       // C: M = v + hi*8, N = lr
    Otile[(long long)(v + hi * 8) * n + lr] = make_float2(cr[v], ci[v]);
  }
}

// ------------------------------------------------------------- MLP (WMMA f32)
// Partial GEMM: part[ks][16][N] = A[16, kchunk] * W[kchunk, N]
__global__ __launch_bounds__(32) void fc_part(float* __restrict__ part,
                                              const float* __restrict__ A,
                                              const float* __restrict__ W,
                                              int K, int N, int kChunk) {
  const int lane = threadIdx.x & 31;
  const int tn = blockIdx.x;
  const int ks = blockIdx.y;
  const int lr = lane & 15;
  const int hi = lane >> 4;
  const int col = tn * 16 + lr;
  int kbeg = ks * kChunk;
  int kend = kbeg + kChunk; if (kend > K) kend = K;
  v8f c = {0.f, 0.f, 0.f, 0.f, 0.f, 0.f, 0.f, 0.f};
  for (int kk = kbeg; kk < kend; kk += 4) {
    int k0 = kk + hi * 2;
    __builtin_prefetch((const void*)(W + (long long)(k0 + 64) * N + col), 0, 1);
    v2f a = {A[lr * K + k0], A[lr * K + k0 + 1]};
    v2f b = {W[(long long)k0 * N + col], W[(long long)(k0 + 1) * N + col]};
    c = wmma_f32x4(a, b, c);
  }
  float* P = part + (long long)ks * 16 * N;
#pragma unroll
  for (int v = 0; v < 8; v++) P[(long long)(v + hi * 8) * N + col] = c[v];
}

__device__ __forceinline__ float bn_relu(float x, float bias, float g, float bb,
                                         float m, float vv) {
  x += bias;
  x = fmaxf(x, 0.0f);
  return (x - m) * rsqrtf(vv + BN_EPS) * g + bb;
}

__global__ void l1_reduce(float* out, const float* part, const float* bias,
                          const float* g, const float* bb, const float* mm,
                          const float* vv, int N, int S) {
  int idx = blockIdx.x * blockDim.x + threadIdx.x;
  if (idx >= BATCH * N) return;
  int col = idx % N, b = idx / N;
  float s = 0.0f;
  for (int k = 0; k < S; k++) s += part[(long long)(k * 16 + b) * N + col];
  out[idx] = bn_relu(s, bias[col], g[col], bb[col], mm[col], vv[col]);
}

// Fused small layer: out[16,N] = bn(relu(A[16,K] * W[K,N] + bias))
__global__ __launch_bounds__(32) void fc_fused(float* __restrict__ out,
                                               const float* __restrict__ A,
                                               const float* __restrict__ W,
                                               const float* bias, const float* g,
                                               const float* bb, const float* mm,
                                               const float* vv, int K, int N) {
  const int lane = threadIdx.x & 31;
  const int tn = blockIdx.x;
  const int lr = lane & 15;
  const int hi = lane >> 4;
  const int col = tn * 16 + lr;
  v8f c = {0.f, 0.f, 0.f, 0.f, 0.f, 0.f, 0.f, 0.f};
  for (int kk = 0; kk < K; kk += 4) {
    int k0 = kk + hi * 2;
    v2f a = {A[lr * K + k0], A[lr * K + k0 + 1]};
    v2f b = {W[(long long)k0 * N + col], W[(long long)(k0 + 1) * N + col]};
    c = wmma_f32x4(a, b, c);
  }
  float bi = bias[col], gg = g[col], bb2 = bb[col], m2 = mm[col], v2 = vv[col];
#pragma unroll
  for (int v = 0; v < 8; v++)
    out[(long long)(v + hi * 8) * N + col] = bn_relu(c[v], bi, gg, bb2, m2, v2);
}

__global__ void fc_final(float* out, const float* A, const float* w, const float* b4) {
  __shared__ float red[128];
  int b = blockIdx.x, t = threadIdx.x;
  red[t] = A[b * 128 + t] * w[t];
  __syncthreads();
  for (int s = 64; s > 0; s >>= 1) {
    if (t < s) red[t] += red[t + s];
    __syncthreads();
  }
  if (t == 0) out[b] = red[0] + b4[0];
}

// ============================================================ host orchestration
static inline unsigned gblk(long long total, int bs) {
  return (unsigned)((total + bs - 1) / bs);
}

static void run_cgemm(float2* out, const float2* A, const float2* W,
                      int Mrows, int n, hipStream_t s) {
  int tilesM = Mrows / 16, tilesN = n / 16;   // tilesM is always a multiple of 8
  cgemm_dft<<<dim3(tilesN, tilesM / 8), 256, 0, s>>>(out, A, W, Mrows, n);
}

static void run_transpose(float2* out, const float2* in, int n, int G, hipStream_t s) {
  ctranspose<<<dim3(n / 16, n / 16, G), dim3(16, 16), 0, s>>>(out, in, n);
}

// io -> fft2(io) using tmp as scratch (two axis GEMMs + transposes)
static void fft2_inplace(float2* io, float2* tmp, const float2* W, int G, int n,
                         hipStream_t s) {
  run_cgemm(tmp, io, W, G * n, n, s);
  run_transpose(io, tmp, n, G, s);
  run_cgemm(tmp, io, W, G * n, n, s);
  run_transpose(io, tmp, n, G, s);
}

extern "C" void kernel_launch(void* const* d_in, const int* in_sizes, int n_in,
                              void* d_out, int out_size, void* d_ws, size_t ws_size,
                              hipStream_t stream) {
  (void)in_sizes; (void)n_in; (void)out_size; (void)ws_size;

  const float* x    = (const float*)d_in[0];
  const float* w1   = (const float*)d_in[1];
  const float* b1   = (const float*)d_in[2];
  const float* bn1g = (const float*)d_in[3];
  const float* bn1b = (const float*)d_in[4];
  const float* bn1m = (const float*)d_in[5];
  const float* bn1v = (const float*)d_in[6];
  const float* w2   = (const float*)d_in[7];
  const float* b2   = (const float*)d_in[8];
  const float* bn2g = (const float*)d_in[9];
  const float* bn2b = (const float*)d_in[10];
  const float* bn2m = (const float*)d_in[11];
  const float* bn2v = (const float*)d_in[12];
  const float* w3   = (const float*)d_in[13];
  const float* b3   = (const float*)d_in[14];
  const float* bn3g = (const float*)d_in[15];
  const float* bn3b = (const float*)d_in[16];
  const float* bn3m = (const float*)d_in[17];
  const float* bn3v = (const float*)d_in[18];
  const float* w4   = (const float*)d_in[19];
  const float* b4   = (const float*)d_in[20];
  float* out = (float*)d_out;

  // ---- workspace carve-up (~230 MB total) ----
  char* base = (char*)d_ws;
  size_t off = 0;
  auto take = [&](size_t bytes) -> void* {
    void* r = base + off;
    off = (off + bytes + 255) & ~(size_t)255;
    return r;
  };
  float2* Wf[4]; float2* Wi[4]; float* phi[4];
  for (int r = 0; r < 4; r++) {
    int n = 256 >> r;
    Wf[r]  = (float2*)take((size_t)n * n * sizeof(float2));
    Wi[r]  = (float2*)take((size_t)n * n * sizeof(float2));
    phi[r] = (float*) take((size_t)n * n * sizeof(float));
  }
  float* psiB[3][4] = {};
  psiB[0][0] = (float*)take((size_t)8 * 256 * 256 * 4);
  psiB[0][1] = (float*)take((size_t)8 * 256 * 256 * 4);
  psiB[0][2] = (float*)take((size_t)8 * 256 * 256 * 4);
  psiB[0][3] = (float*)take((size_t)8 * 256 * 256 * 4);
  psiB[1][1] = (float*)take((size_t)8 * 128 * 128 * 4);
  psiB[1][2] = (float*)take((size_t)8 * 128 * 128 * 4);
  psiB[2][1] = (float*)take((size_t)8 * 64 * 64 * 4);
  float*  mx   = (float*) take(8 * sizeof(float));
  float2* xf   = (float2*)take((size_t)16 * 65536 * sizeof(float2));
  float2* bufA = (float2*)take((size_t)128 * 65536 * sizeof(float2));
  float2* bufB = (float2*)take((size_t)128 * 65536 * sizeof(float2));
  float2* u1f  = (float2*)take((size_t)128 * 65536 * sizeof(float2));
  float*  h    = (float*) take((size_t)BATCH * FEAT * sizeof(float));
  float*  part = (float*) take((size_t)16 * 16 * 1024 * sizeof(float));
  float*  act1 = (float*) take((size_t)BATCH * 1024 * sizeof(float));
  float*  act2 = (float*) take((size_t)BATCH * 512 * sizeof(float));
  float*  act3 = (float*) take((size_t)BATCH * 128 * sizeof(float));

  // ---- deterministic constant generation (DFT mats, phi, psi banks) ----
  for (int r = 0; r < 4; r++) {
    int n = 256 >> r, n2 = n * n;
    gen_dft<<<gblk(n2, 256), 256, 0, stream>>>(Wf[r], n, -1.0f, 1.0f);
    gen_dft<<<gblk(n2, 256), 256, 0, stream>>>(Wi[r], n,  1.0f, 1.0f / (float)n);
    gen_phi<<<gblk(n2, 256), 256, 0, stream>>>(phi[r], n, 4 - r);
  }
  const int bankR[7]  = {0, 0, 0, 0, 1, 1, 2};
  const int bankJS[7] = {0, 1, 2, 3, 1, 2, 1};
  for (int i = 0; i < 7; i++) {
    int r = bankR[i], js = bankJS[i];
    int n = 256 >> r, n2 = n * n;
    fill_f<<<1, 32, 0, stream>>>(mx, 1e-12f, 8);
    gen_psi_raw<<<gblk((long long)8 * n2, 256), 256, 0, stream>>>(psiB[r][js], mx, n, js);
    psi_norm<<<gblk((long long)8 * n2, 256), 256, 0, stream>>>(psiB[r][js], mx, n2);
  }

  // ---- Xf = fft2(x) ----
  r2c<<<gblk((long long)16 * 65536, 256), 256, 0, stream>>>(xf, x, (long long)16 * 65536);
  fft2_inplace(xf, bufA, Wf[0], 16, 256, stream);

  // ---- S0 = real(ifft2(Xf * phi0))[::16, ::16] ----
  cmul_bcast<<<gblk((long long)16 * 65536, 256), 256, 0, stream>>>(
      bufA, xf, phi[0], (long long)16 * 65536, 65536, 1, 0);
  fft2_inplace(bufA, bufB, Wi[0], 16, 256, stream);
  real_sub_store<<<gblk(16 * 256, 256), 256, 0, stream>>>(h, bufA, 256, 16, 16, 1, 0, 1);

  // ---- first + second order ----
  int blk = 0;
  for (int j1 = 0; j1 < 4; j1++) {
    int m1 = 256 >> j1;
    long long n2full = 65536;
    // y = ifft2(Xf[:,None] * psi(0,j1))      [16,8,256,256]
    cmul_bcast<<<gblk((long long)128 * n2full, 256), 256, 0, stream>>>(
        bufA, xf, psiB[0][j1], (long long)128 * n2full, 65536, 8, 1);
    fft2_inplace(bufA, bufB, Wi[0], 128, 256, stream);
    // u1 = |y| subsampled by 2^j1 (stored complex, imag=0)
    cmod_sub<<<gblk((long long)128 * m1 * m1, 256), 256, 0, stream>>>(
        u1f, bufA, 256, 1 << j1, 128);
    // u1f = fft2(u1)
    fft2_inplace(u1f, bufA, Wf[j1], 128, m1, stream);
    // S1[j1] = real(ifft2(u1f * phi[j1]))[::2^(J-j1)]
    cmul_bcast<<<gblk((long long)128 * m1 * m1, 256), 256, 0, stream>>>(
        bufA, u1f, phi[j1], (long long)128 * m1 * m1, m1 * m1, 1, 0);
    fft2_inplace(bufA, bufB, Wi[j1], 128, m1, stream);
    real_sub_store<<<gblk(128 * 256, 256), 256, 0, stream>>>(
        h, bufA, m1, m1 / 16, 128, 8, 1 + 8 * j1, 1);
    // second order
    for (int j2 = j1 + 1; j2 < 4; j2++) {
      int js = j2 - j1, m2 = 256 >> j2;
      const float* bank = psiB[j1][js];
      for (int l2 = 0; l2 < 8; l2++) {
        // y2 = ifft2(u1f * psi(j1,js)[l2])
        cmul_bcast<<<gblk((long long)128 * m1 * m1, 256), 256, 0, stream>>>(
            bufA, u1f, bank + (long long)l2 * m1 * m1, (long long)128 * m1 * m1,
            m1 * m1, 1, 0);
        fft2_inplace(bufA, bufB, Wi[j1], 128, m1, stream);
        // u2 = |y2| subsampled by 2^js
        cmod_sub<<<gblk((long long)128 * m2 * m2, 256), 256, 0, stream>>>(
            bufB, bufA, m1, 1 << js, 128);
        // S2 = real(ifft2(fft2(u2) * phi[j2]))[::2^(J-j2)]
        fft2_inplace(bufB, bufA, Wf[j2], 128, m2, stream);
        cmul_bcast<<<gblk((long long)128 * m2 * m2, 256), 256, 0, stream>>>(
            bufA, bufB, phi[j2], (long long)128 * m2 * m2, m2 * m2, 1, 0);
        fft2_inplace(bufA, bufB, Wi[j2], 128, m2, stream);
        real_sub_store<<<gblk(128 * 256, 256), 256, 0, stream>>>(
            h, bufA, m2, m2 / 16, 128, 8, 33 + 64 * blk + l2, 8);
      }
      blk++;
    }
  }

  // ---- MLP head ----
  // L1: [16,106752] x [106752,1024], 16-way K-split (437 MB weight stream)
  fc_part<<<dim3(64, 16), 32, 0, stream>>>(part, h, w1, FEAT, 1024, FEAT / 16);
  l1_reduce<<<gblk(BATCH * 1024, 256), 256, 0, stream>>>(
      act1, part, b1, bn1g, bn1b, bn1m, bn1v, 1024, 16);
  // L2 / L3 fused epilogue
  fc_fused<<<32, 32, 0, stream>>>(act2, act1, w2, b2, bn2g, bn2b, bn2m, bn2v, 1024, 512);
  fc_fused<<<8, 32, 0, stream>>>(act3, act2, w3, b3, bn3g, bn3b, bn3m, bn3v, 512, 128);
  // L4: [16,128] x [128,1]
  fc_final<<<16, 128, 0, stream>>>(out, act3, w4, b4);
}